// get_model_3143916061235
// MI455X (gfx1250) — compile-verified
//
#include <hip/hip_runtime.h>
#include <hip/hip_bf16.h>

typedef float v2f __attribute__((ext_vector_type(2)));
typedef float v8f __attribute__((ext_vector_type(8)));

static inline int cdiv(int a, int b) { return (a + b - 1) / b; }

// ---------------------------------------------------------------------------
// Transpose [B,3,N] -> [B,N,3]
// ---------------------------------------------------------------------------
__global__ void k_transpose_xyz(const float* __restrict__ in, float* __restrict__ out,
                                int N, int total) {
  int i = blockIdx.x * blockDim.x + threadIdx.x;
  if (i >= total) return;
  int d = i % 3; int rem = i / 3; int n = rem % N; int b = rem / N;
  out[i] = in[((size_t)b * 3 + d) * N + n];
}

// ---------------------------------------------------------------------------
// Farthest point sampling. One block per batch. Matches jnp.argmax first-index
// tie-break semantics; records `far` BEFORE the distance update (as reference).
// ---------------------------------------------------------------------------
__global__ __launch_bounds__(256)
void k_fps(const float* __restrict__ xyz, float* __restrict__ dist,
           int* __restrict__ out_idx, int N, int S) {
  const int b = blockIdx.x;
  const int tid = threadIdx.x;
  const int nt = blockDim.x;
  const float* P = xyz + (size_t)b * N * 3;
  float* D = dist + (size_t)b * N;
  __shared__ float sval[256];
  __shared__ int   sidx[256];
  for (int n = tid; n < N; n += nt) D[n] = 1e10f;
  __syncthreads();
  int far = 0;
  for (int t = 0; t < S; ++t) {
    if (tid == 0) out_idx[b * S + t] = far;
    float cx = P[far * 3 + 0], cy = P[far * 3 + 1], cz = P[far * 3 + 2];
    float best = -1.0f; int bi = N - 1;
    for (int n = tid; n < N; n += nt) {
      float dx = P[n * 3 + 0] - cx, dy = P[n * 3 + 1] - cy, dz = P[n * 3 + 2] - cz;
      float d = dx * dx + dy * dy + dz * dz;
      float dd = fminf(D[n], d);
      D[n] = dd;
      if (dd > best) { best = dd; bi = n; }   // ascending n -> first max kept
    }
    sval[tid] = best; sidx[tid] = bi;
    __syncthreads();
    for (int s2 = nt >> 1; s2 > 0; s2 >>= 1) {
      if (tid < s2) {
        float v2 = sval[tid + s2]; int i2 = sidx[tid + s2];
        if (v2 > sval[tid] || (v2 == sval[tid] && i2 < sidx[tid])) {
          sval[tid] = v2; sidx[tid] = i2;
        }
      }
      __syncthreads();
    }
    far = sidx[0];
    __syncthreads();
  }
}

// ---------------------------------------------------------------------------
// Gather xyz rows by index: out[b,s,:] = xyz[b, idx[b,s], :]
// ---------------------------------------------------------------------------
__global__ void k_gather_xyz(const float* __restrict__ xyz, const int* __restrict__ idx,
                             float* __restrict__ out, int N, int S, int total) {
  int i = blockIdx.x * blockDim.x + threadIdx.x;
  if (i >= total) return;                 // total = B*S*3
  int d = i % 3; int bs = i / 3; int b = bs / S;
  int n = idx[bs];
  out[i] = xyz[((size_t)b * N + n) * 3 + d];
}

// ---------------------------------------------------------------------------
// Ball query: first K indices (ascending) with expanded-form sqdist <= r^2,
// padded with the first hit (reference semantics).
// ---------------------------------------------------------------------------
__global__ void k_ball_query(const float* __restrict__ xyz, const float* __restrict__ new_xyz,
                             int* __restrict__ gi, int N, int S, int K, float r2, int total) {
  int i = blockIdx.x * blockDim.x + threadIdx.x;
  if (i >= total) return;                 // total = B*S
  int b = i / S;
  const float* P = xyz + (size_t)b * N * 3;
  float cx = new_xyz[i * 3 + 0], cy = new_xyz[i * 3 + 1], cz = new_xyz[i * 3 + 2];
  float cc = cx * cx + cy * cy + cz * cz;
  int* G = gi + (size_t)i * K;
  int cnt = 0, first = 0; bool found = false;
  for (int n = 0; n < N; ++n) {
    float px = P[n * 3 + 0], py = P[n * 3 + 1], pz = P[n * 3 + 2];
    float d2 = px * px + py * py + pz * pz + cc - 2.0f * (px * cx + py * cy + pz * cz);
    if (!(d2 > r2)) {
      if (!found) { first = n; found = true; }
      G[cnt++] = n;
      if (cnt >= K) break;
    }
  }
  for (int k = cnt; k < K; ++k) G[k] = first;
}

// ---------------------------------------------------------------------------
// Build grouped features: feat[(b,s,k), 0:C] = pts[b, gi, :],
//                         feat[(b,s,k), C:C+3] = xyz[b, gi, :] - new_xyz[b,s,:]
// ---------------------------------------------------------------------------
__global__ void k_group_concat(const float* __restrict__ pts, const float* __restrict__ xyz,
                               const float* __restrict__ new_xyz, const int* __restrict__ gi,
                               float* __restrict__ feat, int N, int S, int K, int C, int total) {
  int i = blockIdx.x * blockDim.x + threadIdx.x;
  if (i >= total) return;                 // total = B*S*K*(C+3)
  int Cc = C + 3;
  int row = i / Cc; int c = i - row * Cc;
  int b = row / (S * K);
  int s = (row / K) % S;
  int n = gi[row];
  float v;
  if (c < C) v = pts[((size_t)b * N + n) * C + c];
  else {
    int d = c - C;
    v = xyz[((size_t)b * N + n) * 3 + d] - new_xyz[((size_t)b * S + s) * 3 + d];
  }
  feat[i] = v;
}

// ---------------------------------------------------------------------------
// Fused dense + BN + ReLU via V_WMMA_F32_16X16X4_F32.
// X [M,K] row-major, W [N,K] row-major (Y = X*W^T), Y [M,N].
// mode 0: y = relu((x.W^T)*s + (b-m)*s + beta),  s = g*rsqrt(v+eps)
// mode 1: y = x.W^T + b   (classifier head)
// 4 waves/block; each wave owns a 16x32 output strip (one A fragment feeds
// two WMMAs). Main K-loop is fully unconditional: clamped duplicate rows/cols
// only pollute outputs that are never stored (A-row m affects only D-row m,
// B-col n only D-col n). The K-tail zero-fills via clamped load + VALU select
// so EXEC stays all-ones around every WMMA.
// fp32 A=16x4 / B=4x16 fragments: v0 holds K+0 (lanes 0-15) / K+2 (16-31),
// v1 holds K+1 / K+3.
// ---------------------------------------------------------------------------
__global__ __launch_bounds__(128)
void k_gemm_wmma(const float* __restrict__ X, const float* __restrict__ W,
                 const float* __restrict__ bias, const float* __restrict__ g,
                 const float* __restrict__ beta, const float* __restrict__ mean,
                 const float* __restrict__ var, float* __restrict__ Y,
                 int M, int K, int N, int mode) {
  const int lane = threadIdx.x & 31;
  const int wave = threadIdx.x >> 5;
  const int half = lane >> 4;
  const int l = lane & 15;
  const int n0 = blockIdx.x * 32;
  const int m0 = (blockIdx.y * 4 + wave) * 16;

  const int row  = m0 + l;
  const int col0 = n0 + l;
  const int col1 = n0 + 16 + l;
  const int rc  = min(row, M - 1);
  const int cc0 = min(col0, N - 1);
  const int cc1 = min(col1, N - 1);
  const float* xr = X + (size_t)rc * K;
  const float* w0 = W + (size_t)cc0 * K;
  const float* w1 = W + (size_t)cc1 * K;

  v8f acc0 = {}, acc1 = {};
  const int kb = half * 2;                 // lanes 16-31 handle K+2,K+3
  const int Kmain = K & ~3;
  for (int kk = 0; kk < Kmain; kk += 4) {
    const int k0 = kk + kb;
    v2f a, b0, b1;
    a.x  = xr[k0]; a.y  = xr[k0 + 1];
    b0.x = w0[k0]; b0.y = w0[k0 + 1];
    b1.x = w1[k0]; b1.y = w1[k0 + 1];
    acc0 = __builtin_amdgcn_wmma_f32_16x16x4_f32(false, a, false, b0,
                                                 (short)0, acc0, false, false);
    acc1 = __builtin_amdgcn_wmma_f32_16x16x4_f32(false, a, false, b1,
                                                 (short)0, acc1, false, false);
  }
  if (Kmain < K) {
    const int k0 = Kmain + kb, k1 = k0 + 1;
    const int k0c = min(k0, K - 1), k1c = min(k1, K - 1);
    float ax  = xr[k0c], ay  = xr[k1c];
    float b0x = w0[k0c], b0y = w0[k1c];
    float b1x = w1[k0c], b1y = w1[k1c];
    const bool v0 = k0 < K, v1 = k1 < K;
    v2f a, b0, b1;
    a.x  = v0 ? ax  : 0.0f;  a.y  = v1 ? ay  : 0.0f;
    b0.x = v0 ? b0x : 0.0f;  b0.y = v1 ? b0y : 0.0f;
    b1.x = v0 ? b1x : 0.0f;  b1.y = v1 ? b1y : 0.0f;
    acc0 = __builtin_amdgcn_wmma_f32_16x16x4_f32(false, a, false, b0,
                                                 (short)0, acc0, false, false);
    acc1 = __builtin_amdgcn_wmma_f32_16x16x4_f32(false, a, false, b1,
                                                 (short)0, acc1, false, false);
  }

  float scale0, shift0, scale1, shift1;
  if (mode == 0) {
    float s0 = g[cc0] * rsqrtf(var[cc0] + 1e-5f);
    float s1 = g[cc1] * rsqrtf(var[cc1] + 1e-5f);
    scale0 = s0; shift0 = (bias[cc0] - mean[cc0]) * s0 + beta[cc0];
    scale1 = s1; shift1 = (bias[cc1] - mean[cc1]) * s1 + beta[cc1];
  } else {
    scale0 = 1.0f; shift0 = bias[cc0];
    scale1 = 1.0f; shift1 = bias[cc1];
  }
  const bool c0ok = col0 < N;
  const bool c1ok = col1 < N;
#pragma unroll
  for (int i = 0; i < 8; ++i) {
    int r = m0 + i + half * 8;             // C layout: vgpr i -> rows i / i+8
    if (r < M) {
      if (c0ok) {
        float y = acc0[i] * scale0 + shift0;
        if (mode == 0) y = fmaxf(y, 0.0f);
        Y[(size_t)r * N + col0] = y;
      }
      if (c1ok) {
        float y = acc1[i] * scale1 + shift1;
        if (mode == 0) y = fmaxf(y, 0.0f);
        Y[(size_t)r * N + col1] = y;
      }
    }
  }
}

// ---------------------------------------------------------------------------
// Max-pool over neighborhood K; scatter into output channel slice.
// ---------------------------------------------------------------------------
__global__ void k_maxpool(const float* __restrict__ feat, float* __restrict__ out,
                          int K, int C, int outC, int coff, int total) {
  int i = blockIdx.x * blockDim.x + threadIdx.x;
  if (i >= total) return;                 // total = B*S*C
  int c = i % C; int bs = i / C;
  float m = -3.402823466e38f;
  const float* f = feat + ((size_t)bs * K) * C + c;
  for (int k = 0; k < K; ++k) m = fmaxf(m, f[(size_t)k * C]);
  out[(size_t)bs * outC + coff + c] = m;
}

// ---------------------------------------------------------------------------
// 3-NN (expanded-form sqdist, top_k first-index tie semantics) + IDW weights.
// ---------------------------------------------------------------------------
__global__ void k_knn3(const float* __restrict__ xyz1, const float* __restrict__ xyz2,
                       int* __restrict__ idx, float* __restrict__ w,
                       int N1, int N2, int total) {
  int i = blockIdx.x * blockDim.x + threadIdx.x;
  if (i >= total) return;                 // total = B*N1
  int b = i / N1;
  float ax = xyz1[i * 3 + 0], ay = xyz1[i * 3 + 1], az = xyz1[i * 3 + 2];
  float aa = ax * ax + ay * ay + az * az;
  const float* Q = xyz2 + (size_t)b * N2 * 3;
  float d0 = 3.4e38f, d1 = 3.4e38f, d2v = 3.4e38f;
  int i0 = 0, i1 = 0, i2 = 0;
  for (int m = 0; m < N2; ++m) {
    float qx = Q[m * 3 + 0], qy = Q[m * 3 + 1], qz = Q[m * 3 + 2];
    float d = aa + qx * qx + qy * qy + qz * qz - 2.0f * (ax * qx + ay * qy + az * qz);
    if (d < d0)      { d2v = d1; i2 = i1; d1 = d0; i1 = i0; d0 = d; i0 = m; }
    else if (d < d1) { d2v = d1; i2 = i1; d1 = d;  i1 = m; }
    else if (d < d2v){ d2v = d;  i2 = m; }
  }
  float r0 = 1.0f / (d0 + 1e-8f);
  float r1 = 1.0f / (d1 + 1e-8f);
  float r2 = 1.0f / (d2v + 1e-8f);
  float rs = r0 + r1 + r2;
  idx[i * 3 + 0] = i0; idx[i * 3 + 1] = i1; idx[i * 3 + 2] = i2;
  w[i * 3 + 0] = r0 / rs; w[i * 3 + 1] = r1 / rs; w[i * 3 + 2] = r2 / rs;
}

// ---------------------------------------------------------------------------
// X[row, 0:C1] = points1[row]; X[row, C1:C1+C2] = sum_j w_j * points2[idx_j]
// ---------------------------------------------------------------------------
__global__ void k_interp_concat(const float* __restrict__ p1, int C1,
                                const float* __restrict__ p2, int C2,
                                const int* __restrict__ idx, const float* __restrict__ w,
                                float* __restrict__ X, int N1, int N2, int total) {
  int i = blockIdx.x * blockDim.x + threadIdx.x;
  if (i >= total) return;                 // total = B*N1*(C1+C2)
  int C = C1 + C2;
  int row = i / C; int c = i - row * C;
  int b = row / N1;
  float v;
  if (c < C1) {
    v = p1[(size_t)row * C1 + c];
  } else {
    int cc = c - C1;
    const float* base = p2 + (size_t)b * N2 * C2 + cc;
    v = w[row * 3 + 0] * base[(size_t)idx[row * 3 + 0] * C2]
      + w[row * 3 + 1] * base[(size_t)idx[row * 3 + 1] * C2]
      + w[row * 3 + 2] * base[(size_t)idx[row * 3 + 2] * C2];
  }
  X[i] = v;
}

// ---------------------------------------------------------------------------
// Row-wise log_softmax over C (small C=13).
// ---------------------------------------------------------------------------
__global__ void k_log_softmax(const float* __restrict__ in, float* __restrict__ out,
                              int C, int rows) {
  int i = blockIdx.x * blockDim.x + threadIdx.x;
  if (i >= rows) return;
  const float* x = in + (size_t)i * C;
  float m = x[0];
  for (int c = 1; c < C; ++c) m = fmaxf(m, x[c]);
  float s = 0.0f;
  for (int c = 0; c < C; ++c) s += __expf(x[c] - m);
  float lse = m + __logf(s);
  float* y = out + (size_t)i * C;
  for (int c = 0; c < C; ++c) y[c] = x[c] - lse;
}

// ---------------------------------------------------------------------------
// Host orchestration
// ---------------------------------------------------------------------------
struct LayerP { const float *w, *b, *g, *beta, *m, *v; int cin, cout; };

extern "C" void kernel_launch(void* const* d_in, const int* in_sizes, int n_in,
                              void* d_out, int out_size, void* d_ws, size_t ws_size,
                              hipStream_t stream) {
  (void)in_sizes; (void)n_in; (void)out_size; (void)ws_size;
  const int B = 4;
  const int Nl[5] = {16384, 1024, 256, 64, 16};
  const int Cl[5] = {3, 96, 256, 512, 1024};
  const float radii[4][2] = {{0.05f, 0.1f}, {0.1f, 0.2f}, {0.2f, 0.4f}, {0.4f, 0.8f}};
  const int Ksamp[2] = {16, 32};
  const int sa_dims[4][2][3] = {
      {{16, 16, 32},   {32, 32, 64}},
      {{64, 64, 128},  {64, 96, 128}},
      {{128, 196, 256},{128, 196, 256}},
      {{256, 256, 512},{256, 384, 512}},
  };
  const int fp_cin[4]    = {1536, 512, 352, 128};
  const int fp_nlyr[4]   = {2, 2, 2, 3};
  const int fp_dims[4][3] = {{256, 256, 0}, {256, 256, 0}, {256, 128, 0}, {128, 128, 128}};

  // ---- parse parameter pointers (setup_inputs dict insertion order) ----
  int p = 1;  // d_in[0] = xyz
  auto take = [&](int cin, int cout) {
    LayerP L;
    L.w    = (const float*)d_in[p + 0];
    L.b    = (const float*)d_in[p + 1];
    L.g    = (const float*)d_in[p + 2];
    L.beta = (const float*)d_in[p + 3];
    L.m    = (const float*)d_in[p + 4];
    L.v    = (const float*)d_in[p + 5];
    p += 6; L.cin = cin; L.cout = cout;
    return L;
  };
  LayerP sa[4][2][3];
  for (int lvl = 0; lvl < 4; ++lvl)
    for (int sc = 0; sc < 2; ++sc) {
      int cin = Cl[lvl] + 3;
      for (int li = 0; li < 3; ++li) {
        sa[lvl][sc][li] = take(cin, sa_dims[lvl][sc][li]);
        cin = sa_dims[lvl][sc][li];
      }
    }
  LayerP fp[4][3];
  for (int f = 0; f < 4; ++f) {
    int cin = fp_cin[f];
    for (int li = 0; li < fp_nlyr[f]; ++li) {
      fp[f][li] = take(cin, fp_dims[f][li]);
      cin = fp_dims[f][li];
    }
  }
  LayerP head1 = take(128, 128);
  LayerP head2; head2.w = (const float*)d_in[p]; head2.b = (const float*)d_in[p + 1];
  head2.g = head2.beta = head2.m = head2.v = nullptr; head2.cin = 128; head2.cout = 13;

  // ---- workspace bump allocator ----
  size_t off = 0;
  auto alloc_f = [&](size_t n) {
    float* q = (float*)((char*)d_ws + off);
    off += ((n * sizeof(float) + 255) / 256) * 256;
    return q;
  };
  auto alloc_i = [&](size_t n) {
    int* q = (int*)((char*)d_ws + off);
    off += ((n * sizeof(int) + 255) / 256) * 256;
    return q;
  };

  float* xyzL[5];
  xyzL[0] = alloc_f((size_t)B * Nl[0] * 3);     // also level-0 points
  for (int l = 1; l <= 4; ++l) xyzL[l] = alloc_f((size_t)B * Nl[l] * 3);
  float* ptsL[5]; ptsL[0] = xyzL[0];
  for (int l = 1; l <= 4; ++l) ptsL[l] = alloc_f((size_t)B * Nl[l] * Cl[l]);
  float* dist    = alloc_f((size_t)B * Nl[0]);
  float* fpOut0  = alloc_f((size_t)B * 64 * 256);
  float* fpOut1  = alloc_f((size_t)B * 256 * 256);
  float* fpOut2  = alloc_f((size_t)B * 1024 * 128);
  float* knn_w   = alloc_f((size_t)B * Nl[0] * 3);
  float* pingA   = alloc_f((size_t)B * 16384 * 128);   // 8.4M floats
  float* pingB   = alloc_f((size_t)B * 16384 * 128);
  int* fps_idx   = alloc_i((size_t)B * 1024);
  int* gi        = alloc_i((size_t)B * 1024 * 32);
  int* knn_idx   = alloc_i((size_t)B * Nl[0] * 3);

  auto gemm = [&](const float* Xp, const LayerP& L, float* Yp, int M, int mode) {
    dim3 g((unsigned)cdiv(L.cout, 32), (unsigned)cdiv(M, 64));
    k_gemm_wmma<<<g, dim3(128), 0, stream>>>(Xp, L.w, L.b, L.g, L.beta, L.m, L.v,
                                             Yp, M, L.cin, L.cout, mode);
  };

  // ---- l0: transpose xyz to channel-last ----
  {
    int total = B * 3 * Nl[0];
    k_transpose_xyz<<<cdiv(total, 256), 256, 0, stream>>>(
        (const float*)d_in[0], xyzL[0], Nl[0], total);
  }

  // ---- SA levels (multi-scale grouping) ----
  for (int lvl = 0; lvl < 4; ++lvl) {
    const int N = Nl[lvl], S = Nl[lvl + 1], C = Cl[lvl];
    k_fps<<<B, 256, 0, stream>>>(xyzL[lvl], dist, fps_idx, N, S);
    {
      int total = B * S * 3;
      k_gather_xyz<<<cdiv(total, 256), 256, 0, stream>>>(xyzL[lvl], fps_idx,
                                                         xyzL[lvl + 1], N, S, total);
    }
    const float* ptsIn = ptsL[lvl];
    int coff = 0;
    for (int sc = 0; sc < 2; ++sc) {
      const int K = Ksamp[sc];
      const float r = radii[lvl][sc];
      {
        int total = B * S;
        k_ball_query<<<cdiv(total, 128), 128, 0, stream>>>(
            xyzL[lvl], xyzL[lvl + 1], gi, N, S, K, r * r, total);
      }
      {
        int total = B * S * K * (C + 3);
        k_group_concat<<<cdiv(total, 256), 256, 0, stream>>>(
            ptsIn, xyzL[lvl], xyzL[lvl + 1], gi, pingA, N, S, K, C, total);
      }
      const int M = B * S * K;
      const float* src = pingA; float* dst = pingB;
      for (int li = 0; li < 3; ++li) {
        gemm(src, sa[lvl][sc][li], dst, M, 0);
        const float* t = src; src = dst; dst = (float*)t;
      }
      const int Cout = sa_dims[lvl][sc][2];
      {
        int total = B * S * Cout;
        k_maxpool<<<cdiv(total, 256), 256, 0, stream>>>(
            src, ptsL[lvl + 1], K, Cout, Cl[lvl + 1], coff, total);
      }
      coff += Cout;
    }
  }

  // ---- FP (feature propagation) levels ----
  struct FPJob { int N1, N2; const float* p1; int C1; const float* p2; int C2; };
  FPJob jobs[4] = {
      {Nl[3], Nl[4], ptsL[3], 512, ptsL[4], 1024},   // fp0 -> l3p (fpOut0)
      {Nl[2], Nl[3], ptsL[2], 256, fpOut0, 256},     // fp1 -> l2p (fpOut1)
      {Nl[1], Nl[2], ptsL[1], 96,  fpOut1, 256},     // fp2 -> l1p (fpOut2)
      {Nl[0], Nl[1], nullptr, 0,   fpOut2, 128},     // fp3 -> l0p (stays in ping)
  };
  const float* xyz1s[4] = {xyzL[3], xyzL[2], xyzL[1], xyzL[0]};
  const float* xyz2s[4] = {xyzL[4], xyzL[3], xyzL[2], xyzL[1]};
  float* fpDst[3] = {fpOut0, fpOut1, fpOut2};

  const float* l0p = nullptr;
  for (int f = 0; f < 4; ++f) {
    const FPJob& J = jobs[f];
    {
      int total = B * J.N1;
      k_knn3<<<cdiv(total, 128), 128, 0, stream>>>(xyz1s[f], xyz2s[f],
                                                   knn_idx, knn_w, J.N1, J.N2, total);
    }
    {
      int total = B * J.N1 * (J.C1 + J.C2);
      k_interp_concat<<<cdiv(total, 256), 256, 0, stream>>>(
          J.p1, J.C1, J.p2, J.C2, knn_idx, knn_w, pingA, J.N1, J.N2, total);
    }
    const int M = B * J.N1;
    if (fp_nlyr[f] == 2) {
      gemm(pingA, fp[f][0], pingB, M, 0);
      gemm(pingB, fp[f][1], fpDst[f], M, 0);
    } else {
      gemm(pingA, fp[f][0], pingB, M, 0);
      gemm(pingB, fp[f][1], pingA, M, 0);
      gemm(pingA, fp[f][2], pingB, M, 0);
      l0p = pingB;
    }
  }

  // ---- head: conv+bn+relu, classifier, log_softmax ----
  const int Mh = B * Nl[0];
  gemm(l0p, head1, pingA, Mh, 0);
  gemm(pingA, head2, pingB, Mh, 1);
  k_log_softmax<<<cdiv(Mh, 256), 256, 0, stream>>>(pingB, (float*)d_out, 13, Mh);
}